// RoutedCapsuleLayer_15350213116623
// MI455X (gfx1250) — compile-verified
//
#include <hip/hip_runtime.h>

// Capsule dynamic routing, fused + L2-resident-chunked for MI455X (gfx1250).
//
//   u[b,o,i,k] = sum_j W[o,i,k,j] * x[b,i,j]        (WMMA f32 16x16x4, M=batch chunk)
//   3 fused routing scans: blog += v_prev . u ; c = softmax_o(blog) ; s += c * u
//   (scan #1 with v_prev == 0 reproduces the uniform c = 1/64 iteration exactly)
//
// Batch is chunked by 16 so u_chunk (75.5 MB) + W (37.7 MB) stay in the 192 MB L2.
// Workspace layout (floats): [ u_chunk | blog_chunk | s_chunk | v_chunk ]  (~81 MB).

typedef __attribute__((ext_vector_type(2))) float v2f;
typedef __attribute__((ext_vector_type(4))) float v4f;
typedef __attribute__((ext_vector_type(8))) float v8f;

#define N_IN   1152
#define K_INN  8
#define N_OUT  64
#define K_OUT  16
#define B_TOT  64
#define BC     16      // batch chunk (== one WMMA M tile)
#define IG     16      // i-values per scan block

// ---------------------------------------------------------------- zero ----
__global__ __launch_bounds__(256) void zero_k(float* __restrict__ p, int n) {
  int i = blockIdx.x * 256 + threadIdx.x;
  if (i < n) p[i] = 0.f;
}

// ------------------------------------------------------- u generation ----
// One block per i (1152 blocks). 8 waves; wave w computes o-tiles w*8..w*8+7.
// GEMM per i:  [BC=16 x 8] @ [8 x 1024]  ->  [16 x 1024], via 2 chained
// v_wmma_f32_16x16x4_f32 per 16x16 tile (K=0..3 then K=4..7).
//
// A (16x4 f32) lane map: lanes 0-15 hold {K=0,K=1}, lanes 16-31 hold {K=2,K=3},
// M = lane&15.  B (4x16 f32) mirrors with N = lane&15.
// C/D (16x16 f32): VGPR r -> M=r (lanes 0-15, N=lane) / M=r+8 (lanes 16-31).
__global__ __launch_bounds__(256) void gen_u_k(const float* __restrict__ x,
                                               const float* __restrict__ W,
                                               float* __restrict__ u,
                                               int b0) {
  const int i    = blockIdx.x;          // 0..1151
  const int tid  = threadIdx.x;
  const int wave = tid >> 5;            // 0..7
  const int lane = tid & 31;
  const int half = lane >> 4;           // 0 | 1
  const int ln   = lane & 15;           // M (batch row) for A, N (k-col) for B

  // A fragments: x[b0+ln, i, j];  WMMA#1: j = 2*half+{0,1}; WMMA#2: j = 4+2*half+{0,1}
  const float* xp = x + ((size_t)(b0 + ln) * N_IN + i) * K_INN;
  v2f a0 = *(const v2f*)(xp + 2 * half);
  v2f a1 = *(const v2f*)(xp + 4 + 2 * half);

#pragma unroll
  for (int tt = 0; tt < 8; ++tt) {
    const int o = wave * 8 + tt;        // output-capsule tile (N columns o*16..o*16+15)
    // B fragments: B[j][n] = W[o, i, k=ln, j]
    const float* wp = W + (((size_t)o * N_IN + i) * K_OUT + ln) * K_INN;
    v2f bA = *(const v2f*)(wp + 2 * half);
    v2f bB = *(const v2f*)(wp + 4 + 2 * half);

    v8f c = {};
    c = __builtin_amdgcn_wmma_f32_16x16x4_f32(false, a0, false, bA, (short)0, c, false, false);
    c = __builtin_amdgcn_wmma_f32_16x16x4_f32(false, a1, false, bB, (short)0, c, false, false);

    // u layout: u[b_local][i][o][k]
#pragma unroll
    for (int r = 0; r < 8; ++r) {
      const int brow = r + half * 8;    // local batch row (M)
      u[(((size_t)brow * N_IN + i) * N_OUT + o) * K_OUT + ln] = c[r];
    }
  }
}

// ---------------------------------------------------------- fused scan ----
// One block per (b_local, group of IG i's). 256 threads: o = tid>>2, part = tid&3.
// For each i: delta_o = sum_k v_prev.u ; blog += delta ; c = softmax_o ; s += c*u.
__global__ __launch_bounds__(256) void scan_k(const float* __restrict__ u,
                                              float* __restrict__ blog,
                                              const float* __restrict__ vprev,
                                              float* __restrict__ s) {
  const int groups = N_IN / IG;         // 72
  const int bl  = blockIdx.x / groups;
  const int ig  = blockIdx.x % groups;
  const int tid = threadIdx.x;
  const int o    = tid >> 2;            // 0..63
  const int part = tid & 3;             // 4 floats each of K_OUT=16

  __shared__ float sb[N_OUT];
  __shared__ float se[N_OUT];

  const v4f vp = *(const v4f*)(vprev + ((size_t)bl * N_OUT + o) * K_OUT + part * 4);
  v4f acc = {0.f, 0.f, 0.f, 0.f};

  for (int ii = 0; ii < IG; ++ii) {
    const int i = ig * IG + ii;
    const v4f uv = *(const v4f*)(u + (((size_t)bl * N_IN + i) * N_OUT + o) * K_OUT + part * 4);

    // delta_o = <v_prev[o,:], u[o,i,:]>  (4 lanes per o, reduce within wave32)
    float d = uv.x * vp.x + uv.y * vp.y + uv.z * vp.z + uv.w * vp.w;
    d += __shfl_xor(d, 1, 32);
    d += __shfl_xor(d, 2, 32);

    if (part == 0) {
      float* bp = blog + ((size_t)bl * N_OUT + o) * N_IN + i;
      const float bnew = *bp + d;       // scan1: v_prev==0 -> blog stays 0
      *bp = bnew;
      sb[o] = bnew;
    }
    __syncthreads();

    // softmax over the 64 output capsules (LDS broadcast reads)
    float m = -3.402823466e+38f;
    for (int j = 0; j < N_OUT; ++j) m = fmaxf(m, sb[j]);
    if (part == 0) se[o] = __expf(sb[o] - m);
    __syncthreads();
    float sum = 0.f;
    for (int j = 0; j < N_OUT; ++j) sum += se[j];
    const float cc = se[o] / sum;

    acc.x += cc * uv.x; acc.y += cc * uv.y;
    acc.z += cc * uv.z; acc.w += cc * uv.w;
    __syncthreads();                    // protect sb/se for next i
  }

  float* sp = s + ((size_t)bl * N_OUT + o) * K_OUT + part * 4;
  atomicAdd(sp + 0, acc.x);
  atomicAdd(sp + 1, acc.y);
  atomicAdd(sp + 2, acc.z);
  atomicAdd(sp + 3, acc.w);
}

// -------------------------------------------------------------- squash ----
// v = s * n2/(1+n2)/sqrt(n2+eps); also zeroes s for the next scan's atomics.
__global__ __launch_bounds__(256) void squash_k(float* __restrict__ s,
                                                float* __restrict__ v,
                                                float* __restrict__ out,  // nullable
                                                int b0) {
  const int idx = blockIdx.x * 256 + threadIdx.x;   // (b_local * 64 + o)
  if (idx >= BC * N_OUT) return;
  float* sp = s + (size_t)idx * K_OUT;
  float vals[K_OUT];
  float n2 = 0.f;
#pragma unroll
  for (int k = 0; k < K_OUT; ++k) { vals[k] = sp[k]; n2 += vals[k] * vals[k]; }
  const float scale = n2 / (1.f + n2) * rsqrtf(n2 + 1e-7f);
  float* vp = v + (size_t)idx * K_OUT;
#pragma unroll
  for (int k = 0; k < K_OUT; ++k) {
    vp[k] = scale * vals[k];
    sp[k] = 0.f;                        // reset accumulator for next scan
  }
  if (out) {
    const int bl = idx / N_OUT;
    const int o  = idx % N_OUT;
    float* op = out + (((size_t)(b0 + bl)) * N_OUT + o) * K_OUT;
#pragma unroll
    for (int k = 0; k < K_OUT; ++k) op[k] = scale * vals[k];
  }
}

// ---------------------------------------------------------------- host ----
extern "C" void kernel_launch(void* const* d_in, const int* in_sizes, int n_in,
                              void* d_out, int out_size, void* d_ws, size_t ws_size,
                              hipStream_t stream) {
  (void)in_sizes; (void)n_in; (void)out_size; (void)ws_size;
  const float* x = (const float*)d_in[0];   // [64,1152,8]
  const float* W = (const float*)d_in[1];   // [64,1152,16,8]
  float* out = (float*)d_out;               // [64,64,16]

  float* u    = (float*)d_ws;                                   // BC*1152*64*16
  float* blog = u    + (size_t)BC * N_IN * N_OUT * K_OUT;       // BC*64*1152
  float* s    = blog + (size_t)BC * N_OUT * N_IN;               // BC*64*16
  float* v    = s    + (size_t)BC * N_OUT * K_OUT;              // BC*64*16

  const int nzero = BC * N_OUT * N_IN + 2 * BC * N_OUT * K_OUT; // blog + s + v (contiguous)

  for (int c = 0; c < B_TOT / BC; ++c) {
    const int b0 = c * BC;
    zero_k<<<(nzero + 255) / 256, 256, 0, stream>>>(blog, nzero);
    gen_u_k<<<N_IN, 256, 0, stream>>>(x, W, u, b0);
    for (int it = 0; it < 3; ++it) {
      scan_k<<<BC * (N_IN / IG), 256, 0, stream>>>(u, blog, v, s);
      squash_k<<<(BC * N_OUT + 255) / 256, 256, 0, stream>>>(
          s, v, (it == 2) ? out : nullptr, b0);
    }
  }
}